// ParametricRotationMatrix2D_31138512896290
// MI455X (gfx1250) — compile-verified
//
#include <hip/hip_runtime.h>

typedef float v2f __attribute__((ext_vector_type(2)));
typedef float v8f __attribute__((ext_vector_type(8)));

#define N_MAT   32
#define IN_DIM  512
#define HID     1024
#define NANG    496          // 32*31/2
#define MTILE   16           // batch rows per workgroup
#define XPITCH  516          // 512 + 4 pad (bank-conflict-free A loads)
#define HPITCH  1028         // 1024 + 4 pad
#define THREADS 512          // 16 waves (wave32)
#define NWAVES  (THREADS / 32)
#define TWO_PI   6.28318530717958647692f
#define INV_2PI  0.15915494309189533577f

__global__ __launch_bounds__(THREADS)
void fused_rotation_kernel(const float* __restrict__ x,
                           const float* __restrict__ W1,
                           const float* __restrict__ b1,
                           const float* __restrict__ W2,
                           const float* __restrict__ b2,
                           float* __restrict__ out)
{
    // LDS: H tile [16 x 1028] + cs pairs [16 x 496 x 2]; X tile aliases cs area
    // (X is dead after GEMM1, cs is written only in phase 2). ~126 KB total.
    __shared__ __align__(16) float smem[MTILE * HPITCH + MTILE * NANG * 2];
    float* sH  = smem;                       // 16*1028 floats
    float* sCS = smem + MTILE * HPITCH;      // 16*496*2 floats
    float* sX  = sCS;                        // alias: 16*516 floats (phase 1 only)

    const int  tid  = threadIdx.x;
    const int  lane = tid & 31;
    const int  wave = tid >> 5;
    const int  nIdx = lane & 15;            // N (or M) index within the 16x16 tile
    const int  kgrp = (lane >> 4) << 1;     // 0 or 2 : K sub-group per WMMA A/B layout
    const int  mbas = (lane >> 4) << 3;     // 0 or 8 : M base for C/D layout
    const long tile = blockIdx.x;           // which group of 16 batch rows
    const float* xTile = x + tile * (long)MTILE * IN_DIM;

    // ---- Stage X tile into LDS (coalesced) ----
    for (int idx = tid; idx < MTILE * IN_DIM; idx += THREADS) {
        int r = idx >> 9;        // / 512
        int c = idx & 511;
        sX[r * XPITCH + c] = xTile[(long)r * IN_DIM + c];
    }
    __syncthreads();

    // ---- GEMM1: H = relu(X @ W1^T + b1) via V_WMMA_F32_16X16X4_F32 ----
    {
        const int aOff = nIdx * XPITCH + kgrp;        // A fragment base (M row)
        for (int c = wave; c < HID / 16; c += NWAVES) {
            const int n0 = c * 16;
            const float* w1row = W1 + (long)(n0 + nIdx) * IN_DIM + kgrp;
            v8f acc = {};
            #pragma unroll 4
            for (int k = 0; k < IN_DIM; k += 4) {
                v2f a = *(const v2f*)&sX[aOff + k];
                v2f b = *(const v2f*)&w1row[k];
                acc = __builtin_amdgcn_wmma_f32_16x16x4_f32(
                        false, a, false, b, (short)0, acc, false, false);
            }
            const float bias = b1[n0 + nIdx];
            #pragma unroll
            for (int r = 0; r < 8; ++r) {
                float h = acc[r] + bias;
                sH[(mbas + r) * HPITCH + n0 + nIdx] = h > 0.f ? h : 0.f;
            }
        }
    }
    __syncthreads();

    // ---- GEMM2: angles = H @ W2^T + b2 ; then mod 2pi -> (cos,sin) in LDS ----
    {
        const int aOff = nIdx * HPITCH + kgrp;
        for (int c = wave; c < NANG / 16; c += NWAVES) {
            const int n0 = c * 16;
            const float* w2row = W2 + (long)(n0 + nIdx) * HID + kgrp;
            v8f acc = {};
            #pragma unroll 4
            for (int k = 0; k < HID; k += 4) {
                v2f a = *(const v2f*)&sH[aOff + k];
                v2f b = *(const v2f*)&w2row[k];
                acc = __builtin_amdgcn_wmma_f32_16x16x4_f32(
                        false, a, false, b, (short)0, acc, false, false);
            }
            const float bias = b2[n0 + nIdx];
            #pragma unroll
            for (int r = 0; r < 8; ++r) {
                float ang = acc[r] + bias;
                ang = ang - floorf(ang * INV_2PI) * TWO_PI;   // mod 2*pi
                v2f cs;
                cs.x = __cosf(ang);
                cs.y = __sinf(ang);
                const int brow = mbas + r;       // batch row within tile
                const int kAng = n0 + nIdx;      // angle index 0..495
                *(v2f*)&sCS[(brow * NANG + kAng) * 2] = cs;
            }
        }
    }
    __syncthreads();

    // ---- Givens chain: R[b] = prod_k G(i_k, j_k, theta_k), per (b,row) thread ----
    {
        const int brow = tid >> 5;   // batch row in tile (wave-uniform -> LDS broadcast)
        const int row  = lane;       // row of R, 0..31
        float r[N_MAT];
        #pragma unroll
        for (int c = 0; c < N_MAT; ++c) r[c] = (c == row) ? 1.f : 0.f;

        const v2f* cs = (const v2f*)&sCS[brow * NANG * 2];
        int kA = 0;
        #pragma unroll
        for (int i = 0; i < N_MAT - 1; ++i) {
            #pragma unroll
            for (int j = i + 1; j < N_MAT; ++j) {
                v2f p = cs[kA++];                 // (cos, sin) broadcast from LDS
                float ti = r[i], tj = r[j];
                r[i] = fmaf(p.x, ti,  p.y * tj);  //  c*ti + s*tj
                r[j] = fmaf(p.x, tj, -p.y * ti);  // -s*ti + c*tj
            }
        }

        float* dst = out + ((tile * MTILE + brow) * (long)N_MAT + row) * N_MAT;
        #pragma unroll
        for (int c = 0; c < N_MAT; c += 4) {
            float4 v = make_float4(r[c], r[c + 1], r[c + 2], r[c + 3]);
            *(float4*)&dst[c] = v;
        }
    }
}

extern "C" void kernel_launch(void* const* d_in, const int* in_sizes, int n_in,
                              void* d_out, int out_size, void* d_ws, size_t ws_size,
                              hipStream_t stream) {
    const float* x  = (const float*)d_in[0];
    const float* W1 = (const float*)d_in[1];
    const float* b1 = (const float*)d_in[2];
    const float* W2 = (const float*)d_in[3];
    const float* b2 = (const float*)d_in[4];
    float* out = (float*)d_out;

    dim3 grid(16384 / MTILE);   // 1024 workgroups
    dim3 block(THREADS);        // 512 threads = 16 waves
    hipLaunchKernelGGL(fused_rotation_kernel, grid, block, 0, stream,
                       x, W1, b1, W2, b2, out);
}